// ResidualRNN_88656714924676
// MI455X (gfx1250) — compile-verified
//
#include <hip/hip_runtime.h>
#include <hip/hip_bf16.h>

// Residual ReLU-RNN, B=64 T=512 I=128 H=512 L=20 C=10, all fp32.
// Bandwidth/latency bound (~7.7 GB moved vs 0.69 TFLOP) -> keep fp32 and use
// V_WMMA_F32_16X16X4_F32. Whh staging into LDS uses the Tensor Data Mover
// (tensor_load_to_lds + s_wait_tensorcnt) with LDS row padding for
// bank-conflict-free ds_load fragments.

typedef __attribute__((ext_vector_type(2))) float v2f;
typedef __attribute__((ext_vector_type(8))) float v8f;
typedef __attribute__((ext_vector_type(4))) unsigned int su4;
typedef __attribute__((ext_vector_type(8))) unsigned int su8;

__device__ __forceinline__ v8f wmma_f32(v2f a, v2f b, v8f c) {
  // D = A(16x4 f32) x B(4x16 f32) + C(16x16 f32)
  return __builtin_amdgcn_wmma_f32_16x16x4_f32(
      /*neg_a=*/false, a, /*neg_b=*/false, b,
      /*c_mod=*/(short)0, c, /*reuse_a=*/false, /*reuse_b=*/false);
}

// ---------------------------------------------------------------------------
// XT = In(MxK) @ W(NxK)^T + (bih + bhh)          (M = B*T, N = H)
// Wave tile: 32(M) x 32(N) -> 4 loads per 4 WMMAs (vs 5:4 for 16x64).
// Block: 8 waves arranged 4(M) x 2(N) -> 128(M) x 64(N).
// A/B fragments: lane r=lane&15, k = kb + 2*(lane>>4) + {0,1} -> v2f loads.
// ---------------------------------------------------------------------------
__global__ __launch_bounds__(256) void gemm_xt_kernel(
    const float* __restrict__ In, const float* __restrict__ W,
    const float* __restrict__ bih, const float* __restrict__ bhh,
    float* __restrict__ XT, int M, int N, int K)
{
  const int tid  = threadIdx.x;
  const int wave = tid >> 5;
  const int lane = tid & 31;
  const int lrow = lane & 15;
  const int lhi  = lane >> 4;

  const int mwave = blockIdx.x * 128 + (wave & 3) * 32;
  const int nwave = blockIdx.y * 64 + (wave >> 2) * 32;

  v8f c00 = {}, c01 = {}, c10 = {}, c11 = {};

  const float* A0 = In + (size_t)(mwave +      lrow) * K + 2 * lhi;
  const float* A1 = In + (size_t)(mwave + 16 + lrow) * K + 2 * lhi;
  const float* B0 = W  + (size_t)(nwave +      lrow) * K + 2 * lhi;
  const float* B1 = W  + (size_t)(nwave + 16 + lrow) * K + 2 * lhi;

#pragma unroll 2
  for (int kb = 0; kb < K; kb += 4) {
    v2f a0 = *(const v2f*)(A0 + kb);
    v2f a1 = *(const v2f*)(A1 + kb);
    v2f b0 = *(const v2f*)(B0 + kb);
    v2f b1 = *(const v2f*)(B1 + kb);
    c00 = wmma_f32(a0, b0, c00);
    c01 = wmma_f32(a0, b1, c01);
    c10 = wmma_f32(a1, b0, c10);
    c11 = wmma_f32(a1, b1, c11);
  }

  const float bias0 = bih[nwave +      lrow] + bhh[nwave +      lrow];
  const float bias1 = bih[nwave + 16 + lrow] + bhh[nwave + 16 + lrow];

#pragma unroll
  for (int r = 0; r < 8; ++r) {
    // C/D layout: vgpr r -> row r (lanes 0-15) / row r+8 (lanes 16-31)
    float* r0 = XT + (size_t)(mwave +      r + 8 * lhi) * N + nwave + lrow;
    float* r1 = XT + (size_t)(mwave + 16 + r + 8 * lhi) * N + nwave + lrow;
    r0[0] = c00[r] + bias0;   r0[16] = c01[r] + bias1;
    r1[0] = c10[r] + bias0;   r1[16] = c11[r] + bias1;
  }
}

// ---------------------------------------------------------------------------
// Sequential scan: h_t = relu(xt_t + h_{t-1} @ Whh^T); Out(+)= h_t.
// XT (B x T x H) holds xt on entry and is overwritten in place with h.
// 32 persistent blocks: block -> 16-column slice of H; 4 waves -> 16-row
// batch tiles (B=64).
//
// Whh slice (16 rows x 512 K, row-major) is DMA'd into LDS by the Tensor
// Data Mover with padding: pad_interval=7 (every 256 DWORDs) + pad_amount
// code 1 (2 DWORDs) -> LDS row stride 516 floats. 516 % 64 = 4 banks/row,
// so the per-step ds_load_b32 fragment (16 rows x 2 K-phases across 32
// lanes) is bank-conflict-free. Elements with k>=256 sit at +2 floats.
// ---------------------------------------------------------------------------
#define SW_STRIDE 516

__global__ __launch_bounds__(128) void rnn_scan_kernel(
    float* __restrict__ XT, const float* __restrict__ Whh,
    float* __restrict__ Out, unsigned* __restrict__ bar,
    int layer, int T, int H, int addResidual)
{
  __shared__ float sW[16 * SW_STRIDE];           // 16 rows, padded stride

  const int tid  = threadIdx.x;
  const int wave = tid >> 5;
  const int lane = tid & 31;
  const int lrow = lane & 15;
  const int lhi  = lane >> 4;

  const int nbase = blockIdx.x * 16;             // output-column slice
  const int mbase = wave * 16;                   // batch-row tile

  if (tid < 32) {
    // ---- Tensor DMA descriptor (ISA ch.8): 2D tile, 2 SGPR groups ----
    const unsigned long long ga =
        (unsigned long long)(uintptr_t)(const void*)(Whh + (size_t)nbase * H);
    su4 g0;
    g0.x = 1u;                                   // count=1, user mode
    g0.y = (unsigned)(uintptr_t)(void*)sW;       // lds_addr (byte offset)
    g0.z = (unsigned)(ga & 0xFFFFFFFFu);         // global_addr[31:0]
    g0.w = (unsigned)((ga >> 32) & 0x01FFFFFFu)  // global_addr[56:32]
         | (2u << 30);                           // type = 2 ("image")
    su8 g1;
    g1.s0 = (2u << 16)                           // data_size = 4 bytes
          | (1u << 20)                           // pad_enable
          | (7u << 22)                           // pad_interval: 256 DWORDs
          | (1u << 25);                          // pad_amount: 2 DWORDs
    g1.s1 = (512u & 0xFFFFu) << 16;              // tensor_dim0 = 512 (lo16)
    g1.s2 = (16u << 16);                         // tensor_dim1 = 16
    g1.s3 = (512u << 16);                        // tile_dim0 = 512
    g1.s4 = 16u;                                 // tile_dim1 = 16
    g1.s5 = (unsigned)H;                         // tensor_dim0_stride = 512
    g1.s6 = 0u;
    g1.s7 = 0u;
    asm volatile("tensor_load_to_lds %0, %1" :: "s"(g0), "s"(g1) : "memory");
    __builtin_amdgcn_s_wait_tensorcnt(0);
  }
  __syncthreads();

  const unsigned nblocks = gridDim.x;
  const float* wrow = sW + lrow * SW_STRIDE;     // this lane's Whh row

  for (int t = 0; t < T; ++t) {
    v8f accA = {}, accB = {};
    if (t > 0) {
      const float* hprev =
          XT + ((size_t)(mbase + lrow) * T + (t - 1)) * H + 2 * lhi;
      // two independent WMMA chains (K halves) for ILP on the critical path
#pragma unroll 4
      for (int kb = 0; kb < 256; kb += 4) {
        v2f a = *(const v2f*)(hprev + kb);
        v2f b;
        b.x = wrow[kb + 2 * lhi];
        b.y = wrow[kb + 2 * lhi + 1];
        accA = wmma_f32(a, b, accA);
      }
#pragma unroll 4
      for (int kb = 256; kb < 512; kb += 4) {
        v2f a = *(const v2f*)(hprev + kb);
        v2f b;
        b.x = wrow[kb + 2 * lhi + 2];            // +2: intra-row LDS pad
        b.y = wrow[kb + 2 * lhi + 3];
        accB = wmma_f32(a, b, accB);
      }
    }

#pragma unroll
    for (int r = 0; r < 8; ++r) {
      const int m = mbase + r + 8 * lhi;         // batch index
      const size_t idx = ((size_t)m * T + t) * H + nbase + lrow;
      float h = accA[r] + accB[r] + XT[idx];     // + xt_t
      h = fmaxf(h, 0.0f);
      XT[idx] = h;                               // h sequence, feeds t+1
      if (addResidual) Out[idx] += h; else Out[idx] = h;
    }

    // device-wide barrier between time steps (monotonic target: no resets)
    __threadfence();
    __syncthreads();
    if (tid == 0) {
      __hip_atomic_fetch_add(bar, 1u, __ATOMIC_RELEASE, __HIP_MEMORY_SCOPE_AGENT);
      const unsigned target = (unsigned)(layer * T + t + 1) * nblocks;
      while (__hip_atomic_load(bar, __ATOMIC_ACQUIRE, __HIP_MEMORY_SCOPE_AGENT)
             < target) {
        __builtin_amdgcn_s_sleep(2);
      }
    }
    __syncthreads();
  }
}

// ---------------------------------------------------------------------------
// logits = Out[:, T-1, :] @ fcW^T + fcb   (64 x 10, negligible)
// ---------------------------------------------------------------------------
__global__ void fc_kernel(const float* __restrict__ Out,
                          const float* __restrict__ fcW,
                          const float* __restrict__ fcb,
                          float* __restrict__ y, int T, int H, int C)
{
  const int b = blockIdx.x;
  const int c = threadIdx.x;
  if (c >= C) return;
  const float* last = Out + ((size_t)b * T + (T - 1)) * H;
  float s = fcb[c];
  for (int k = 0; k < H; ++k) s += last[k] * fcW[(size_t)c * H + k];
  y[b * C + c] = s;
}

// ---------------------------------------------------------------------------
extern "C" void kernel_launch(void* const* d_in, const int* in_sizes, int n_in,
                              void* d_out, int out_size, void* d_ws, size_t ws_size,
                              hipStream_t stream)
{
  (void)in_sizes; (void)n_in; (void)out_size; (void)ws_size;
  const float* x    = (const float*)d_in[0];   // (B,T,I)
  const float* Wih0 = (const float*)d_in[1];   // (H,I)
  const float* Wih  = (const float*)d_in[2];   // (L-1,H,H)
  const float* Whh  = (const float*)d_in[3];   // (L,H,H)
  const float* bih  = (const float*)d_in[4];   // (L,H)
  const float* bhh  = (const float*)d_in[5];   // (L,H)
  const float* fcW  = (const float*)d_in[6];   // (C,H)
  const float* fcb  = (const float*)d_in[7];   // (C)
  float* out = (float*)d_out;                  // (B,C)

  const int B = 64, T = 512, I = 128, H = 512, L = 20, C = 10;
  const size_t seq = (size_t)B * T * H;        // 16 Mi floats = 64 MB

  float*    XT  = (float*)d_ws;                // [0, 64MB)
  float*    Out = XT + seq;                    // [64MB, 128MB)
  unsigned* bar = (unsigned*)(Out + seq);      // barrier counter

  hipMemsetAsync(bar, 0, sizeof(unsigned), stream);

  const int M = B * T;
  for (int l = 0; l < L; ++l) {
    const float* In = (l == 0) ? x : Out;
    const float* W  = (l == 0) ? Wih0 : (Wih + (size_t)(l - 1) * H * H);
    const int    K  = (l == 0) ? I : H;

    gemm_xt_kernel<<<dim3(M / 128, H / 64), dim3(256), 0, stream>>>(
        In, W, bih + (size_t)l * H, bhh + (size_t)l * H, XT, M, H, K);

    rnn_scan_kernel<<<dim3(H / 16), dim3(128), 0, stream>>>(
        XT, Whh + (size_t)l * H * H, Out, bar, l, T, H, (l > 0) ? 1 : 0);
  }

  fc_kernel<<<dim3(B), dim3(32), 0, stream>>>(Out, fcW, fcb, out, T, H, C);
}